// BasicNet4_88184268521954
// MI455X (gfx1250) — compile-verified
//
#include <hip/hip_runtime.h>
#include <hip/hip_bf16.h>

// ---------------------------------------------------------------------------
// Emformer-style block for MI455X (gfx1250, wave32).
// Dense GEMMs: register-blocked V_WMMA_F32_16X16X4_F32, 64x64 output tile per
// wave (16 accumulators), 16 FLOP/byte from L2 -> matrix-pipe bound.
// Attention is 5x5 per segment -> plain VALU (~0.4% of FLOPs).
// ---------------------------------------------------------------------------

#define B_    8
#define T_    1025
#define D_    512
#define H_    4
#define DH_   128
#define L_    4
#define U_    1024
#define R_    256
#define NSEG_ 256
#define ROWS_ 1280                       // R_ + U_ tokens per batch
#define MROWS (B_ * ROWS_)               // 10240 GEMM rows
#define GEMM_WAVES 2                     // waves (64-row supertiles) per block

typedef __attribute__((ext_vector_type(2))) float v2f;
typedef __attribute__((ext_vector_type(8))) float v8f;

// ---------------------------------------------------------------------------
// cat = concat(r, u):  row j<256 -> x[:, (j+1)*4],  row j>=256 -> x[:, j-256]
// ---------------------------------------------------------------------------
__global__ void build_cat_kernel(const float* __restrict__ X, float* __restrict__ Cat)
{
    size_t i = (size_t)blockIdx.x * blockDim.x + threadIdx.x;
    const size_t total = (size_t)MROWS * D_;
    if (i >= total) return;
    int d = (int)(i % D_);
    size_t rb = i / D_;
    int j = (int)(rb % ROWS_);
    int b = (int)(rb / ROWS_);
    int src = (j < R_) ? ((j + 1) * 4) : (j - R_);   // rc_idx[255] = 1024 = 256*4
    Cat[i] = X[((size_t)b * T_ + src) * D_ + d];
}

// ---------------------------------------------------------------------------
// LayerNorm over D=512, one wave32 per row (16 elements per lane).
// ---------------------------------------------------------------------------
__global__ void ln_kernel(const float* __restrict__ X, const float* __restrict__ g,
                          const float* __restrict__ bta, float* __restrict__ Y, int rows)
{
    int lane = threadIdx.x & 31;
    int wave = threadIdx.x >> 5;
    int row  = blockIdx.x * (blockDim.x >> 5) + wave;
    if (row >= rows) return;

    const float* x = X + (size_t)row * D_;
    float vals[16];
    float s = 0.f;
#pragma unroll
    for (int i = 0; i < 16; ++i) { vals[i] = x[lane + 32 * i]; s += vals[i]; }
#pragma unroll
    for (int off = 16; off > 0; off >>= 1) s += __shfl_xor(s, off, 32);
    float mu = s * (1.f / (float)D_);

    float vs = 0.f;
#pragma unroll
    for (int i = 0; i < 16; ++i) { float d = vals[i] - mu; vs += d * d; }
#pragma unroll
    for (int off = 16; off > 0; off >>= 1) vs += __shfl_xor(vs, off, 32);
    float inv = rsqrtf(vs * (1.f / (float)D_) + 1e-5f);

    float* y = Y + (size_t)row * D_;
#pragma unroll
    for (int i = 0; i < 16; ++i) {
        int c = lane + 32 * i;
        y[c] = (vals[i] - mu) * inv * g[c] + bta[c];
    }
}

// ---------------------------------------------------------------------------
// C[m,n] = sum_k A[m,k] * W[n,k] + bias[n] (+ residual[m,n]) (ReLU optional)
// A: MxK row-major, W: NxK row-major (i.e. act @ W.T), C: MxN.
// Each wave owns a 64x64 output supertile: 4x4 grid of 16x16 WMMA tiles.
// Per K-step of 4: 8 b64 loads, 16 V_WMMA_F32_16X16X4_F32 -> 16 FLOP/B.
// Frag layouts (f32): A/B lane = (m|n)=L&15, regs = K {0,1}|{2,3} by lane
// half; C/D lane col = L&15, rows = 8*(L>>4)+r in reg r.
// ---------------------------------------------------------------------------
__global__ void __launch_bounds__(GEMM_WAVES * 32)
gemm_wmma_f32_kernel(const float* __restrict__ A,
                     const float* __restrict__ W,
                     const float* __restrict__ bias,
                     const float* __restrict__ residual,
                     float* __restrict__ C,
                     int M, int N, int K, int relu)
{
    int lane = threadIdx.x & 31;
    int wib  = threadIdx.x >> 5;
    int tM   = blockIdx.y * GEMM_WAVES + wib;
    if (tM * 64 >= M) return;                    // wave-uniform: EXEC stays full
    int m0 = tM * 64;
    int n0 = blockIdx.x * 64;

    int lo    = lane & 15;
    int hi    = lane >> 4;
    int kHalf = hi * 2;                          // 0 or 2

    const float* aP[4];
    const float* wP[4];
#pragma unroll
    for (int i = 0; i < 4; ++i) {
        aP[i] = A + (size_t)(m0 + 16 * i + lo) * K + kHalf;
        wP[i] = W + (size_t)(n0 + 16 * i + lo) * K + kHalf;
    }

    v8f acc[4][4];
    const v8f zero = {};
#pragma unroll
    for (int i = 0; i < 4; ++i)
#pragma unroll
        for (int j = 0; j < 4; ++j) acc[i][j] = zero;

    for (int k = 0; k < K; k += 4) {
        v2f a[4], b[4];
#pragma unroll
        for (int i = 0; i < 4; ++i) a[i] = *(const v2f*)(aP[i] + k);  // 8B aligned
#pragma unroll
        for (int j = 0; j < 4; ++j) b[j] = *(const v2f*)(wP[j] + k);
#pragma unroll
        for (int i = 0; i < 4; ++i)
#pragma unroll
            for (int j = 0; j < 4; ++j)
                acc[i][j] = __builtin_amdgcn_wmma_f32_16x16x4_f32(
                    /*neg_a=*/false, a[i], /*neg_b=*/false, b[j],
                    /*c_mod=*/(short)0, acc[i][j],
                    /*reuse_a=*/false, /*reuse_b=*/false);
    }

#pragma unroll
    for (int i = 0; i < 4; ++i) {
        int rowBase = m0 + 16 * i + hi * 8;
#pragma unroll
        for (int j = 0; j < 4; ++j) {
            int col = n0 + 16 * j + lo;
            float bv = bias ? bias[col] : 0.f;
#pragma unroll
            for (int r = 0; r < 8; ++r) {
                size_t idx = (size_t)(rowBase + r) * N + col;
                float v = acc[i][j][r] + bv;
                if (residual) v += residual[idx];
                if (relu) v = fmaxf(v, 0.f);
                C[idx] = v;
            }
        }
    }
}

// ---------------------------------------------------------------------------
// Segment attention: segment s = rows {s, 256+4s..256+4s+3} (5 tokens).
// One block (32 threads) per (b, s); thread t<20 handles (query q=t%5, head
// h=t/5): 5 dots of length 128, softmax over 5, weighted sum of V.
// ---------------------------------------------------------------------------
__global__ void seg_attn_kernel(const float* __restrict__ Q,
                                const float* __restrict__ KV,
                                float* __restrict__ Out)
{
    int bs = blockIdx.x;
    int b  = bs / NSEG_;
    int s  = bs % NSEG_;
    int t  = threadIdx.x;
    if (t >= 20) return;
    int qi = t % 5;
    int h  = t / 5;

    int rows[5];
    rows[0] = s;
#pragma unroll
    for (int j = 0; j < 4; ++j) rows[1 + j] = R_ + 4 * s + j;

    const float scale = 0.088388347648318447f;   // 128^-0.5
    const float* q = Q + ((size_t)b * ROWS_ + rows[qi]) * D_ + h * DH_;

    float sc[5];
#pragma unroll
    for (int j = 0; j < 5; ++j) {
        const float* k = KV + ((size_t)b * ROWS_ + rows[j]) * (2 * D_) + h * DH_;
        float d = 0.f;
        for (int c = 0; c < DH_; c += 4) {
            d += q[c]     * k[c]     + q[c + 1] * k[c + 1]
               + q[c + 2] * k[c + 2] + q[c + 3] * k[c + 3];
        }
        sc[j] = d * scale;
    }
    float mx = sc[0];
#pragma unroll
    for (int j = 1; j < 5; ++j) mx = fmaxf(mx, sc[j]);
    float den = 0.f;
#pragma unroll
    for (int j = 0; j < 5; ++j) { sc[j] = __expf(sc[j] - mx); den += sc[j]; }
    float rden = 1.f / den;
#pragma unroll
    for (int j = 0; j < 5; ++j) sc[j] *= rden;

    const float* v0 = KV + ((size_t)b * ROWS_ + rows[0]) * (2 * D_) + D_ + h * DH_;
    const float* v1 = KV + ((size_t)b * ROWS_ + rows[1]) * (2 * D_) + D_ + h * DH_;
    const float* v2 = KV + ((size_t)b * ROWS_ + rows[2]) * (2 * D_) + D_ + h * DH_;
    const float* v3 = KV + ((size_t)b * ROWS_ + rows[3]) * (2 * D_) + D_ + h * DH_;
    const float* v4 = KV + ((size_t)b * ROWS_ + rows[4]) * (2 * D_) + D_ + h * DH_;
    float* out = Out + ((size_t)b * ROWS_ + rows[qi]) * D_ + h * DH_;
    for (int c = 0; c < DH_; ++c) {
        out[c] = sc[0] * v0[c] + sc[1] * v1[c] + sc[2] * v2[c]
               + sc[3] * v3[c] + sc[4] * v4[c];
    }
}

// ---------------------------------------------------------------------------
// out[b,d] = mean over the 1024 u-rows of cat.  Threads over d -> coalesced.
// ---------------------------------------------------------------------------
__global__ void mean_u_kernel(const float* __restrict__ Cat, float* __restrict__ Out)
{
    int d = blockIdx.x * blockDim.x + threadIdx.x;
    int b = blockIdx.y;
    if (d >= D_) return;
    const float* p = Cat + ((size_t)b * ROWS_ + R_) * D_ + d;
    float s = 0.f;
    for (int r = 0; r < U_; ++r) s += p[(size_t)r * D_];
    Out[b * D_ + d] = s * (1.f / (float)U_);
}

// ---------------------------------------------------------------------------
static inline void launch_gemm(const float* A, const float* W, const float* bias,
                               const float* res, float* C, int M, int N, int K,
                               int relu, hipStream_t stream)
{
    dim3 block(GEMM_WAVES * 32);
    dim3 grid(N / 64, (M / 64) / GEMM_WAVES);    // M=10240 -> 160 supertiles, /2 exact
    gemm_wmma_f32_kernel<<<grid, block, 0, stream>>>(A, W, bias, res, C, M, N, K, relu);
}

extern "C" void kernel_launch(void* const* d_in, const int* in_sizes, int n_in,
                              void* d_out, int out_size, void* d_ws, size_t ws_size,
                              hipStream_t stream)
{
    const float* x       = (const float*)d_in[0];
    const float* Wq      = (const float*)d_in[1];
    const float* bq      = (const float*)d_in[2];
    const float* Wkv     = (const float*)d_in[3];
    const float* bkv     = (const float*)d_in[4];
    const float* Wo      = (const float*)d_in[5];
    const float* bo      = (const float*)d_in[6];
    const float* W1      = (const float*)d_in[7];
    const float* b1      = (const float*)d_in[8];
    const float* W2      = (const float*)d_in[9];
    const float* b2      = (const float*)d_in[10];
    const float* ln_in_g = (const float*)d_in[11];
    const float* ln_in_b = (const float*)d_in[12];
    const float* ff_ln_g = (const float*)d_in[13];
    const float* ff_ln_b = (const float*)d_in[14];
    const float* ln_out_g= (const float*)d_in[15];
    const float* ln_out_b= (const float*)d_in[16];
    float* out = (float*)d_out;

    // Workspace carve-up (floats): total = MROWS*(4*512 + 128 + 1024) ~ 125 MiB
    float* ws     = (float*)d_ws;
    float* catBuf = ws;                                   // MROWS*512
    float* lnBuf  = catBuf + (size_t)MROWS * D_;          // MROWS*512 (also attn out)
    float* qBuf   = lnBuf  + (size_t)MROWS * D_;          // MROWS*512 (also ffn2 out)
    float* rcBuf  = qBuf   + (size_t)MROWS * D_;          // MROWS*512
    float* hBuf   = rcBuf  + (size_t)MROWS * D_;          // MROWS*128
    float* kvBuf  = hBuf   + (size_t)MROWS * DH_;         // MROWS*1024

    // cat = [r, u]
    {
        size_t total = (size_t)MROWS * D_;
        build_cat_kernel<<<(unsigned)((total + 255) / 256), 256, 0, stream>>>(x, catBuf);
    }

    const int lnBlocks = MROWS / 8;                       // 256 thr = 8 waves = 8 rows

    for (int l = 0; l < L_; ++l) {
        const float* Wq_l  = Wq  + (size_t)l * D_ * D_;
        const float* Wkv_l = Wkv + (size_t)l * 2 * D_ * D_;
        const float* Wo_l  = Wo  + (size_t)l * D_ * D_;
        const float* W1_l  = W1  + (size_t)l * 128 * D_;
        const float* W2_l  = W2  + (size_t)l * D_ * 128;

        // ln = LN(cat)
        ln_kernel<<<lnBlocks, 256, 0, stream>>>(catBuf, ln_in_g + l * D_, ln_in_b + l * D_,
                                                lnBuf, MROWS);
        // q = ln @ Wq.T + bq ; kv = ln @ Wkv.T + bkv
        launch_gemm(lnBuf, Wq_l,  bq  + l * D_,     nullptr, qBuf,  MROWS, D_,     D_, 0, stream);
        launch_gemm(lnBuf, Wkv_l, bkv + l * 2 * D_, nullptr, kvBuf, MROWS, 2 * D_, D_, 0, stream);
        // segment attention (q,kv) -> lnBuf (ln fully consumed above)
        seg_attn_kernel<<<B_ * NSEG_, 32, 0, stream>>>(qBuf, kvBuf, lnBuf);
        // rc = attn @ Wo.T + bo + cat
        launch_gemm(lnBuf, Wo_l, bo + l * D_, catBuf, rcBuf, MROWS, D_, D_, 0, stream);
        // ffh = LN(rc)  (into catBuf, which is now free)
        ln_kernel<<<lnBlocks, 256, 0, stream>>>(rcBuf, ff_ln_g + l * D_, ff_ln_b + l * D_,
                                                catBuf, MROWS);
        // h = relu(ffh @ W1.T + b1)
        launch_gemm(catBuf, W1_l, b1 + l * 128, nullptr, hBuf, MROWS, 128, D_, 1, stream);
        // rc2 = h @ W2.T + b2 + rc
        launch_gemm(hBuf, W2_l, b2 + l * D_, rcBuf, qBuf, MROWS, D_, 128, 0, stream);
        // cat_{l+1} = LN(rc2)
        ln_kernel<<<lnBlocks, 256, 0, stream>>>(qBuf, ln_out_g + l * D_, ln_out_b + l * D_,
                                                catBuf, MROWS);
    }

    // out[b,:] = mean over u-rows
    {
        dim3 grid(D_ / 256, B_);
        mean_u_kernel<<<grid, 256, 0, stream>>>(catBuf, out);
    }
}